// AttentivePooling_61555471286328
// MI455X (gfx1250) — compile-verified
//
#include <hip/hip_runtime.h>
#include <stdint.h>

// ---------------------------------------------------------------------------
// AttentivePooling, single pass over h_all (memory-bound: 256 MiB @ 23.3 TB/s
// => ~11.5 us floor). Scores via chained V_WMMA_F32_16X16X4_F32 (full f32
// precision), tiles staged with global_load_async_to_lds_b128 (ASYNCcnt DMA).
// ---------------------------------------------------------------------------

namespace {
constexpr int T_DIM = 2048;
constexpr int B_DIM = 64;
constexpr int D_DIM = 512;
constexpr int ROW_STRIDE = D_DIM + 4;                 // +4 floats pad: 516*4 % 256 == 16 -> 4-bank row skew
constexpr int WAVES = 4;
constexpr int WG_THREADS = WAVES * 32;                // wave32
constexpr int ROWS_PER_WAVE = 16;
constexpr int ROWS_PER_WG = WAVES * ROWS_PER_WAVE;    // 64 timesteps per workgroup
constexpr int N_TBLK = T_DIM / ROWS_PER_WG;           // 32
constexpr float INV_SQRT_D = 0.04419417382415922f;    // 1/sqrt(512)
}

typedef __attribute__((ext_vector_type(2))) float v2f;
typedef __attribute__((ext_vector_type(8))) float v8f;

__device__ __forceinline__ uint32_t lds_off(const void* p) {
  // Low 32 bits of a generic pointer to LDS are the LDS byte address (ISA 10.2).
  return (uint32_t)(uintptr_t)p;
}

__device__ __forceinline__ void async_g2l_b128(uint32_t lds_byte_addr, const float* gaddr) {
  asm volatile("global_load_async_to_lds_b128 %0, %1, off"
               :: "v"(lds_byte_addr), "v"(gaddr)
               : "memory");
}

__device__ __forceinline__ void wait_async0() {
  asm volatile("s_wait_asynccnt 0" ::: "memory");
}

__global__ void zero_f32(float* __restrict__ p, int n) {
  const int i = blockIdx.x * blockDim.x + threadIdx.x;
  if (i < n) p[i] = 0.0f;
}

__global__ void __launch_bounds__(WG_THREADS)
attn_pool_partial(const float* __restrict__ h_all,
                  float* __restrict__ s_part,     // [B, D] accumulators
                  float* __restrict__ z_part)     // [B]   accumulators
{
  extern __shared__ float smem[];
  float* tiles  = smem;                                          // WAVES*16*ROW_STRIDE
  float* vlast  = tiles + WAVES * ROWS_PER_WAVE * ROW_STRIDE;    // [512] h_last[b,:]
  float* outacc = vlast + D_DIM;                                 // [512] per-WG partial out
  float* zacc   = outacc + D_DIM;                                // [1]

  const int tid  = threadIdx.x;
  const int lane = tid & 31;
  const int wave = tid >> 5;
  const int b    = blockIdx.x & (B_DIM - 1);
  const int tblk = blockIdx.x >> 6;
  const int t0   = tblk * ROWS_PER_WG + wave * ROWS_PER_WAVE;

  for (int i = tid; i < D_DIM; i += WG_THREADS) outacc[i] = 0.0f;
  if (tid == 0) zacc[0] = 0.0f;

  // ---- Stage h_last[b,:]: one float4 per thread via async DMA.
  {
    const float* g = h_all + ((size_t)(T_DIM - 1) * B_DIM + b) * D_DIM + tid * 4;
    async_g2l_b128(lds_off(vlast + tid * 4), g);
  }

  // ---- Stage this wave's 16x512 row tile via async DMA (coalesced b128).
  float* mytile = tiles + wave * ROWS_PER_WAVE * ROW_STRIDE;
#pragma unroll
  for (int r = 0; r < ROWS_PER_WAVE; ++r) {
    const float* grow = h_all + ((size_t)(t0 + r) * B_DIM + b) * D_DIM + lane * 4;
    const uint32_t l0 = lds_off(mytile + r * ROW_STRIDE + lane * 4);
#pragma unroll
    for (int q = 0; q < 4; ++q)
      async_g2l_b128(l0 + (uint32_t)(q * 512), grow + q * 128);
  }
  wait_async0();
  __syncthreads();   // vlast contributions come from all waves

  // ---- Scores for 16 rows: C = A(16x512) * v, chained 16x16x4 f32 WMMA.
  // A layout (ISA 7.12.2): lane<16 -> (M=lane, K=0,1); lane>=16 -> (M=lane-16, K=2,3).
  // B is v replicated across all 16 columns -> every C column = score vector.
  const int hh = lane >> 4;          // K half select
  const int rA = lane & 15;          // A row
  const float* arow = mytile + rA * ROW_STRIDE + 2 * hh;
  const float* bcol = vlast + 2 * hh;
  v8f c0 = {0,0,0,0,0,0,0,0};
  v8f c1 = {0,0,0,0,0,0,0,0};
#pragma unroll 8
  for (int k = 0; k < D_DIM; k += 8) {
    v2f a0 = *(const v2f*)(arow + k);
    v2f b0 = *(const v2f*)(bcol + k);
    c0 = __builtin_amdgcn_wmma_f32_16x16x4_f32(false, a0, false, b0,
                                               (short)0, c0, false, false);
    v2f a1 = *(const v2f*)(arow + k + 4);
    v2f b1 = *(const v2f*)(bcol + k + 4);
    c1 = __builtin_amdgcn_wmma_f32_16x16x4_f32(false, a1, false, b1,
                                               (short)0, c1, false, false);
  }

  // C layout: VGPR j holds M=j (lanes 0-15) / M=j+8 (lanes 16-31). Exchange
  // halves so every lane owns all 16 relu'd, scaled scores.
  float sc[16];
  float zpart = 0.0f;
#pragma unroll
  for (int j = 0; j < 8; ++j) {
    float mine  = c0[j] + c1[j];
    float other = __shfl_xor(mine, 16, 32);
    float lowm  = (lane < 16) ? mine : other;   // rows 0..7
    float highm = (lane < 16) ? other : mine;   // rows 8..15
    sc[j]     = fmaxf(lowm,  0.0f) * INV_SQRT_D;
    sc[j + 8] = fmaxf(highm, 0.0f) * INV_SQRT_D;
  }
#pragma unroll
  for (int j = 0; j < 16; ++j) zpart += sc[j];
  if (lane == 0) atomicAdd(zacc, zpart);        // once per 16-row tile

  // ---- Weighted pooling: acc[d] += sc[r] * tile[r][d]; lane owns d = q*128 + lane*4.
  float4 acc[4];
#pragma unroll
  for (int q = 0; q < 4; ++q) acc[q] = make_float4(0.f, 0.f, 0.f, 0.f);
#pragma unroll
  for (int r = 0; r < ROWS_PER_WAVE; ++r) {
    const float w = sc[r];
    const float* row = mytile + r * ROW_STRIDE + lane * 4;
#pragma unroll
    for (int q = 0; q < 4; ++q) {
      const float4 hv = *(const float4*)(row + q * 128);
      acc[q].x = fmaf(w, hv.x, acc[q].x);
      acc[q].y = fmaf(w, hv.y, acc[q].y);
      acc[q].z = fmaf(w, hv.z, acc[q].z);
      acc[q].w = fmaf(w, hv.w, acc[q].w);
    }
  }
#pragma unroll
  for (int q = 0; q < 4; ++q) {
    const int d = q * 128 + lane * 4;
    atomicAdd(&outacc[d + 0], acc[q].x);
    atomicAdd(&outacc[d + 1], acc[q].y);
    atomicAdd(&outacc[d + 2], acc[q].z);
    atomicAdd(&outacc[d + 3], acc[q].w);
  }
  __syncthreads();

  // ---- Flush workgroup partials to global accumulators.
  for (int i = tid; i < D_DIM; i += WG_THREADS)
    atomicAdd(&s_part[(size_t)b * D_DIM + i], outacc[i]);
  if (tid == 0) atomicAdd(&z_part[b], zacc[0]);
}

__global__ void attn_pool_finalize(const float* __restrict__ s_part,
                                   const float* __restrict__ z_part,
                                   float* __restrict__ out) {
  const int i = blockIdx.x * blockDim.x + threadIdx.x;   // i < B*D
  const int b = i / D_DIM;
  out[i] = s_part[i] / (z_part[b] + 1e-9f);
}

extern "C" void kernel_launch(void* const* d_in, const int* in_sizes, int n_in,
                              void* d_out, int out_size, void* d_ws, size_t ws_size,
                              hipStream_t stream) {
  (void)in_sizes; (void)n_in; (void)out_size; (void)ws_size;
  const float* h_all = (const float*)d_in[0];   // [T, B, D] fp32; d_in[1] (xin) unused by reference
  float* out    = (float*)d_out;                // [B, D] fp32
  float* s_part = (float*)d_ws;                 // B*D floats
  float* z_part = s_part + B_DIM * D_DIM;       // B floats

  const int nacc = B_DIM * D_DIM + B_DIM;
  zero_f32<<<dim3((nacc + 255) / 256), dim3(256), 0, stream>>>(s_part, nacc);

  const size_t shmem =
      (size_t)(WAVES * ROWS_PER_WAVE * ROW_STRIDE + 2 * D_DIM + 1) * sizeof(float);
  attn_pool_partial<<<dim3(B_DIM * N_TBLK), dim3(WG_THREADS), shmem, stream>>>(
      h_all, s_part, z_part);

  attn_pool_finalize<<<dim3((B_DIM * D_DIM) / 256), dim3(256), 0, stream>>>(
      s_part, z_part, out);
}